// TransformerOCRHead_59605556134626
// MI455X (gfx1250) — compile-verified
//
#include <hip/hip_runtime.h>
#include <math.h>

// ---------------------------------------------------------------------------
// TransformerOCRHead forward for MI455X (gfx1250), wave32 + WMMA.
// Dense projections use V_WMMA_F32_16X16X4_F32 (full fp32 precision).
// Each wave produces a 16(M) x 64(N) strip: the A fragment is reused by
// 4 independent WMMA accumulator chains (4x less A traffic, latency hiding).
// Weights (~185MB fp32 for the GEMMs, incl. W_out) are L2-resident (192MB L2); the next K-chunk
// is pulled ahead with global_prefetch_b8.
// ---------------------------------------------------------------------------

typedef float v2f __attribute__((ext_vector_type(2)));
typedef float v8f __attribute__((ext_vector_type(8)));

#define EDIM 512
#define HEADS 8
#define DH 64
#define TT 50
#define BB 4
#define NN 32
#define VV 10000
#define FFD 2048
#define BN 128        // B*N
#define MDEC 6400     // B*N*T

// ---------------- WMMA GEMM: C = act(A[M,K] @ W[N,K]^T + bias) -------------
// A-matrix lane layout (16x4 f32): lane = m + 16*half
//   a.x = A[m][k0 + 2*half], a.y = A[m][k0 + 2*half + 1]
// B-matrix lane layout (4x16 f32): lane = n + 16*half
//   b.x = W[n][k0 + 2*half], b.y = W[n][k0 + 2*half + 1]   (W row-major [N,K])
// C/D (16x16 f32): VGPR i -> row m0 + i + 8*half, col n0 + (lane&15)
template<int NT, int RELU>
__global__ __launch_bounds__(32)
void gemm_wmma_kernel(const float* __restrict__ A, const float* __restrict__ W,
                      const float* __restrict__ bias, float* __restrict__ C,
                      int M, int Nld, int K, int n_base) {
  const int n0   = n_base + blockIdx.x * (16 * NT);
  const int m0   = blockIdx.y * 16;
  const int lane = threadIdx.x;
  const int half = lane >> 4;
  const int r    = lane & 15;

  const float* arow = A + (size_t)(m0 + r) * K + 2 * half;
  const float* wrow[NT];
#pragma unroll
  for (int j = 0; j < NT; ++j)
    wrow[j] = W + (size_t)(n0 + 16 * j + r) * K + 2 * half;

  v8f acc[NT] = {};
  for (int k0 = 0; k0 < K; k0 += 4) {
    if ((k0 & 28) == 0 && k0 + 32 < K) {      // prefetch next 128B chunk
      __builtin_prefetch(arow + k0 + 32, 0, 1);
#pragma unroll
      for (int j = 0; j < NT; ++j) __builtin_prefetch(wrow[j] + k0 + 32, 0, 1);
    }
    v2f a = *(const v2f*)(arow + k0);
#pragma unroll
    for (int j = 0; j < NT; ++j) {
      v2f b = *(const v2f*)(wrow[j] + k0);
      acc[j] = __builtin_amdgcn_wmma_f32_16x16x4_f32(
          /*neg_a=*/false, a, /*neg_b=*/false, b,
          /*c_mod=*/(short)0, acc[j], /*reuse_a=*/false, /*reuse_b=*/false);
    }
  }

#pragma unroll
  for (int j = 0; j < NT; ++j) {
    const float bv = bias ? bias[n0 + 16 * j + r] : 0.0f;
#pragma unroll
    for (int i = 0; i < 8; ++i) {
      float v = acc[j][i] + bv;
      if (RELU) v = fmaxf(v, 0.0f);
      C[(size_t)(m0 + i + 8 * half) * Nld + n0 + 16 * j + r] = v;
    }
  }
}

// ---------------- residual add + LayerNorm (row = 512) ---------------------
// x[row] = LN(x[row] + delta[row/ddiv]) * g + b   (ddiv=TT broadcasts cross-attn)
__global__ __launch_bounds__(128)
void add_ln_kernel(float* __restrict__ x, const float* __restrict__ delta,
                   const float* __restrict__ g, const float* __restrict__ b,
                   int ddiv) {
  __shared__ float ls[128], ls2[128];
  const int row = blockIdx.x;
  const int tid = threadIdx.x;
  float* xr = x + (size_t)row * EDIM;
  const float* dr = delta + (size_t)(ddiv == 1 ? row : row / ddiv) * EDIM;

  float v[4];
  float s = 0.0f, s2 = 0.0f;
#pragma unroll
  for (int i = 0; i < 4; ++i) {
    const int e = tid + i * 128;
    const float t = xr[e] + dr[e];
    v[i] = t; s += t; s2 += t * t;
  }
  ls[tid] = s; ls2[tid] = s2; __syncthreads();
  for (int off = 64; off > 0; off >>= 1) {
    if (tid < off) { ls[tid] += ls[tid + off]; ls2[tid] += ls2[tid + off]; }
    __syncthreads();
  }
  const float mean = ls[0] * (1.0f / EDIM);
  const float var  = ls2[0] * (1.0f / EDIM) - mean * mean;
  const float inv  = rsqrtf(var + 1e-5f);
#pragma unroll
  for (int i = 0; i < 4; ++i) {
    const int e = tid + i * 128;
    xr[e] = (v[i] - mean) * inv * g[e] + b[e];
  }
}

// ---------------- context mean over 8 embeddings ---------------------------
__global__ void ctxmean_kernel(const float* __restrict__ ctx, float* __restrict__ cm) {
  const int i = blockIdx.x * blockDim.x + threadIdx.x;
  if (i >= BB * EDIM) return;
  const int b = i >> 9, e = i & 511;
  float s = 0.0f;
  for (int j = 0; j < 8; ++j) s += ctx[((size_t)b * 8 + j) * EDIM + e];
  cm[i] = s * 0.125f;
}

// ---------------- x += pos[n] + ctxmean[b] ---------------------------------
__global__ void pos_ctx_kernel(float* __restrict__ x, const float* __restrict__ pos,
                               const float* __restrict__ cm) {
  const int i = blockIdx.x * blockDim.x + threadIdx.x;
  if (i >= BN * EDIM) return;
  const int tok = i >> 9, e = i & 511;
  const int b = tok >> 5, n = tok & 31;
  x[i] += pos[(size_t)n * EDIM + e] + cm[(size_t)b * EDIM + e];
}

// ---------------- encoder self-attention (N=32, dh=64) ---------------------
// one wave per (b,h); lane = query index n
__global__ __launch_bounds__(32)
void enc_attn_kernel(const float* __restrict__ qkv, float* __restrict__ o) {
  const int b = blockIdx.x >> 3;
  const int h = blockIdx.x & 7;
  const int n = threadIdx.x;
  const float* base = qkv + (size_t)b * NN * (3 * EDIM);
  const float* qp = base + (size_t)n * (3 * EDIM) + h * DH;

  float s[NN];
  float m = -1e30f;
  for (int kk = 0; kk < NN; ++kk) {
    const float* kp = base + (size_t)kk * (3 * EDIM) + EDIM + h * DH;
    float d = 0.0f;
    for (int e = 0; e < DH; ++e) d += qp[e] * kp[e];
    s[kk] = d * 0.125f;   // 1/sqrt(64)
    m = fmaxf(m, s[kk]);
  }
  float sum = 0.0f;
  for (int kk = 0; kk < NN; ++kk) { s[kk] = expf(s[kk] - m); sum += s[kk]; }
  const float is = 1.0f / sum;
  for (int d = 0; d < DH; ++d) {
    float acc = 0.0f;
    for (int kk = 0; kk < NN; ++kk)
      acc += s[kk] * base[(size_t)kk * (3 * EDIM) + 2 * EDIM + h * DH + d];
    o[(size_t)(b * NN + n) * EDIM + h * DH + d] = acc * is;
  }
}

// ---------------- decoder self-attention (T=50, dh=64, per region) ---------
__global__ __launch_bounds__(64)
void dec_attn_kernel(const float* __restrict__ qkv, float* __restrict__ o) {
  const int bn = blockIdx.x >> 3;
  const int h  = blockIdx.x & 7;
  const int t  = threadIdx.x;
  if (t >= TT) return;
  const float* base = qkv + (size_t)bn * TT * (3 * EDIM);
  const float* qp = base + (size_t)t * (3 * EDIM) + h * DH;

  float s[TT];
  float m = -1e30f;
  for (int kk = 0; kk < TT; ++kk) {
    const float* kp = base + (size_t)kk * (3 * EDIM) + EDIM + h * DH;
    float d = 0.0f;
    for (int e = 0; e < DH; ++e) d += qp[e] * kp[e];
    s[kk] = d * 0.125f;
    m = fmaxf(m, s[kk]);
  }
  float sum = 0.0f;
  for (int kk = 0; kk < TT; ++kk) { s[kk] = expf(s[kk] - m); sum += s[kk]; }
  const float is = 1.0f / sum;
  for (int d = 0; d < DH; ++d) {
    float acc = 0.0f;
    for (int kk = 0; kk < TT; ++kk)
      acc += s[kk] * base[(size_t)kk * (3 * EDIM) + 2 * EDIM + h * DH + d];
    o[(size_t)(bn * TT + t) * EDIM + h * DH + d] = acc * is;
  }
}

// ---------------- char embedding gather ------------------------------------
__global__ void gather_kernel(const int* __restrict__ idx, const float* __restrict__ emb,
                              float* __restrict__ y) {
  const int i = blockIdx.x * blockDim.x + threadIdx.x;
  if (i >= MDEC * EDIM) return;
  const int row = i >> 9, e = i & 511;
  y[i] = emb[(size_t)idx[row] * EDIM + e];
}

// ---------------- det head second layer (N=4, tiny) ------------------------
__global__ void det_boxes_kernel(const float* __restrict__ bh, const float* __restrict__ w2,
                                 const float* __restrict__ b2, float* __restrict__ out) {
  const int i = blockIdx.x * blockDim.x + threadIdx.x;
  if (i >= BN * 4) return;
  const int row = i >> 2, j = i & 3;
  const float* hr = bh + (size_t)row * 256;
  const float* wr = w2 + (size_t)j * 256;
  float s = b2[j];
  for (int k = 0; k < 256; ++k) s += hr[k] * wr[k];
  out[i] = s;
}

// ---------------- max(softmax) per logit row: 1/sum(exp(l-m)) --------------
__global__ __launch_bounds__(256)
void pmax_kernel(const float* __restrict__ logits, float* __restrict__ pm) {
  __shared__ float red[256];
  const int row = blockIdx.x;
  const int tid = threadIdx.x;
  const float* lr = logits + (size_t)row * VV;
  float m = -1e30f;
  for (int v = tid; v < VV; v += 256) m = fmaxf(m, lr[v]);
  red[tid] = m; __syncthreads();
  for (int off = 128; off > 0; off >>= 1) {
    if (tid < off) red[tid] = fmaxf(red[tid], red[tid + off]);
    __syncthreads();
  }
  m = red[0]; __syncthreads();
  float s = 0.0f;
  for (int v = tid; v < VV; v += 256) s += expf(lr[v] - m);
  red[tid] = s; __syncthreads();
  for (int off = 128; off > 0; off >>= 1) {
    if (tid < off) red[tid] += red[tid + off];
    __syncthreads();
  }
  if (tid == 0) pm[row] = 1.0f / red[0];
}

__global__ void scores_kernel(const float* __restrict__ pm, float* __restrict__ out) {
  const int i = threadIdx.x;
  if (i >= BN) return;
  float s = 0.0f;
  for (int t = 0; t < TT; ++t) s += pm[i * TT + t];
  out[i] = s * (1.0f / TT);
}

__global__ void copy_kernel(const float* __restrict__ src, float* __restrict__ dst, int n) {
  const int i = blockIdx.x * blockDim.x + threadIdx.x;
  if (i < n) dst[i] = src[i];
}

// ---------------------------------------------------------------------------
static inline void gemm(const float* A, const float* W, const float* bias, float* C,
                        int M, int N, int K, bool relu, hipStream_t s) {
  const int n64 = (N / 64) * 64;                 // wide strips (16x64 per wave)
  if (n64 > 0) {
    dim3 g(n64 / 64, M / 16);
    if (relu) gemm_wmma_kernel<4, 1><<<g, dim3(32), 0, s>>>(A, W, bias, C, M, N, K, 0);
    else      gemm_wmma_kernel<4, 0><<<g, dim3(32), 0, s>>>(A, W, bias, C, M, N, K, 0);
  }
  if (N - n64 > 0) {                             // 16-wide tail (only V=10000)
    dim3 g((N - n64) / 16, M / 16);
    if (relu) gemm_wmma_kernel<1, 1><<<g, dim3(32), 0, s>>>(A, W, bias, C, M, N, K, n64);
    else      gemm_wmma_kernel<1, 0><<<g, dim3(32), 0, s>>>(A, W, bias, C, M, N, K, n64);
  }
}

extern "C" void kernel_launch(void* const* d_in, const int* in_sizes, int n_in,
                              void* d_out, int out_size, void* d_ws, size_t ws_size,
                              hipStream_t stream) {
  (void)in_sizes; (void)n_in; (void)out_size; (void)ws_size;

  // Inputs flattened in setup_inputs() dict (insertion) order, recursively.
  const float* features = (const float*)d_in[0];   // [4,32,256]
  const float* ctx      = (const float*)d_in[1];   // [4,8,512]
  const int*   ttext    = (const int*)  d_in[2];   // [4,32,50]
  const float* W_in     = (const float*)d_in[3];   // [512,256]
  const float* b_in     = (const float*)d_in[4];
  const float* pos      = (const float*)d_in[5];   // [1,500,512]
  // encoder stack (leading L=6 dim on each)
  const float* e_qkv_w = (const float*)d_in[6];   const float* e_qkv_b = (const float*)d_in[7];
  const float* e_out_w = (const float*)d_in[8];   const float* e_out_b = (const float*)d_in[9];
  const float* e_ff1_w = (const float*)d_in[10];  const float* e_ff1_b = (const float*)d_in[11];
  const float* e_ff2_w = (const float*)d_in[12];  const float* e_ff2_b = (const float*)d_in[13];
  const float* e_ln1_g = (const float*)d_in[14];  const float* e_ln1_b = (const float*)d_in[15];
  const float* e_ln2_g = (const float*)d_in[16];  const float* e_ln2_b = (const float*)d_in[17];
  // decoder stack
  const float* d_qkv_w = (const float*)d_in[18];  const float* d_qkv_b = (const float*)d_in[19];
  const float* d_aout_w= (const float*)d_in[20];  const float* d_aout_b= (const float*)d_in[21];
  const float* d_ff1_w = (const float*)d_in[22];  const float* d_ff1_b = (const float*)d_in[23];
  const float* d_ff2_w = (const float*)d_in[24];  const float* d_ff2_b = (const float*)d_in[25];
  const float* d_ln1_g = (const float*)d_in[26];  const float* d_ln1_b = (const float*)d_in[27];
  const float* d_ln2_g = (const float*)d_in[28];  const float* d_ln2_b = (const float*)d_in[29];
  const float* d_caqkv_w=(const float*)d_in[30];  const float* d_caqkv_b=(const float*)d_in[31];
  const float* d_caout_w=(const float*)d_in[32];  const float* d_caout_b=(const float*)d_in[33];
  const float* d_ln3_g = (const float*)d_in[34];  const float* d_ln3_b = (const float*)d_in[35];
  const float* char_emb= (const float*)d_in[36];  // [10000,512]
  const float* W_out   = (const float*)d_in[37];  // [10000,512]
  const float* b_out   = (const float*)d_in[38];
  const float* det_w1  = (const float*)d_in[39];  const float* det_b1 = (const float*)d_in[40];
  const float* det_w2  = (const float*)d_in[41];  const float* det_b2 = (const float*)d_in[42];

  float* out = (float*)d_out;
  const size_t O_BOX = 64000000;   // after text_logits [4,32,50,10000]
  const size_t O_SCR = 64000512;   // after text_boxes  [4,32,4]
  const size_t O_ENC = 64000640;   // after text_scores [4,32]

  // workspace layout (floats)
  float* ws   = (float*)d_ws;
  float* FFH  = ws;                            // 6400*2048
  float* QKV  = FFH + (size_t)MDEC * FFD;      // 6400*1536
  float* Y    = QKV + (size_t)MDEC * 3 * EDIM; // 6400*512
  float* S1   = Y   + (size_t)MDEC * EDIM;     // 6400*512 (attn out)
  float* S2   = S1  + (size_t)MDEC * EDIM;     // 6400*512 (proj scratch)
  float* X    = S2  + (size_t)MDEC * EDIM;     // 128*512 (encoder state)
  float* CM   = X   + (size_t)BN * EDIM;       // 4*512
  float* CV   = CM  + (size_t)BB * EDIM;       // 128*512
  float* CO   = CV  + (size_t)BN * EDIM;       // 128*512
  float* BH   = CO  + (size_t)BN * EDIM;       // 128*256
  float* PM   = BH  + (size_t)BN * 256;        // 6400

  // --- input projection + pos + ctx-mean ---
  ctxmean_kernel<<<(BB * EDIM + 255) / 256, 256, 0, stream>>>(ctx, CM);
  gemm(features, W_in, b_in, X, BN, EDIM, 256, false, stream);
  pos_ctx_kernel<<<(BN * EDIM + 255) / 256, 256, 0, stream>>>(X, pos, CM);

  // --- encoder (post-norm) ---
  for (int i = 0; i < 6; ++i) {
    gemm(X, e_qkv_w + (size_t)i * 3 * EDIM * EDIM, e_qkv_b + (size_t)i * 3 * EDIM,
         QKV, BN, 3 * EDIM, EDIM, false, stream);
    enc_attn_kernel<<<BB * HEADS, 32, 0, stream>>>(QKV, S1);
    gemm(S1, e_out_w + (size_t)i * EDIM * EDIM, e_out_b + (size_t)i * EDIM,
         S2, BN, EDIM, EDIM, false, stream);
    add_ln_kernel<<<BN, 128, 0, stream>>>(X, S2, e_ln1_g + (size_t)i * EDIM,
                                          e_ln1_b + (size_t)i * EDIM, 1);
    gemm(X, e_ff1_w + (size_t)i * FFD * EDIM, e_ff1_b + (size_t)i * FFD,
         FFH, BN, FFD, EDIM, true, stream);
    gemm(FFH, e_ff2_w + (size_t)i * EDIM * FFD, e_ff2_b + (size_t)i * EDIM,
         S2, BN, EDIM, FFD, false, stream);
    add_ln_kernel<<<BN, 128, 0, stream>>>(X, S2, e_ln2_g + (size_t)i * EDIM,
                                          e_ln2_b + (size_t)i * EDIM, 1);
  }

  // --- detection head ---
  gemm(X, det_w1, det_b1, BH, BN, 256, EDIM, true, stream);
  det_boxes_kernel<<<(BN * 4 + 255) / 256, 256, 0, stream>>>(BH, det_w2, det_b2, out + O_BOX);

  // --- decoder init: char embedding gather ---
  gather_kernel<<<(MDEC * EDIM + 255) / 256, 256, 0, stream>>>(ttext, char_emb, Y);

  // --- decoder (post-norm, cross-attn over memory length 1 == broadcast) ---
  for (int i = 0; i < 6; ++i) {
    gemm(Y, d_qkv_w + (size_t)i * 3 * EDIM * EDIM, d_qkv_b + (size_t)i * 3 * EDIM,
         QKV, MDEC, 3 * EDIM, EDIM, false, stream);
    dec_attn_kernel<<<BN * HEADS, 64, 0, stream>>>(QKV, S1);
    gemm(S1, d_aout_w + (size_t)i * EDIM * EDIM, d_aout_b + (size_t)i * EDIM,
         S2, MDEC, EDIM, EDIM, false, stream);
    add_ln_kernel<<<MDEC, 128, 0, stream>>>(Y, S2, d_ln1_g + (size_t)i * EDIM,
                                            d_ln1_b + (size_t)i * EDIM, 1);
    // cross-attention: softmax over 1 key == 1 -> out = (mem@Wv^T+bv)@Wo^T+bo
    gemm(X, d_caqkv_w + (size_t)i * 3 * EDIM * EDIM + (size_t)2 * EDIM * EDIM,
         d_caqkv_b + (size_t)i * 3 * EDIM + 2 * EDIM, CV, BN, EDIM, EDIM, false, stream);
    gemm(CV, d_caout_w + (size_t)i * EDIM * EDIM, d_caout_b + (size_t)i * EDIM,
         CO, BN, EDIM, EDIM, false, stream);
    add_ln_kernel<<<MDEC, 128, 0, stream>>>(Y, CO, d_ln2_g + (size_t)i * EDIM,
                                            d_ln2_b + (size_t)i * EDIM, TT);
    gemm(Y, d_ff1_w + (size_t)i * FFD * EDIM, d_ff1_b + (size_t)i * FFD,
         FFH, MDEC, FFD, EDIM, true, stream);
    gemm(FFH, d_ff2_w + (size_t)i * EDIM * FFD, d_ff2_b + (size_t)i * EDIM,
         S2, MDEC, EDIM, FFD, false, stream);
    add_ln_kernel<<<MDEC, 128, 0, stream>>>(Y, S2, d_ln3_g + (size_t)i * EDIM,
                                            d_ln3_b + (size_t)i * EDIM, 1);
  }

  // --- vocabulary projection (dominant GEMM, 65 GFLOP) straight into d_out ---
  gemm(Y, W_out, b_out, out, MDEC, VV, EDIM, false, stream);

  // --- text_scores = mean_t max_v softmax = mean_t 1/sum(exp(l-m)) ---
  pmax_kernel<<<MDEC, 256, 0, stream>>>(out, PM);
  scores_kernel<<<1, 128, 0, stream>>>(PM, out + O_SCR);

  // --- encoded output ---
  copy_kernel<<<(BN * EDIM + 255) / 256, 256, 0, stream>>>(X, out + O_ENC, BN * EDIM);
}